// RecurrentScanGLRU_86560770883915
// MI455X (gfx1250) — compile-verified
//
#include <hip/hip_runtime.h>

#define D_H   1024
#define TTOT  4096
#define CH3   3072
#define NB    8
#define NCHUNK 16
#define TCH   256          // timesteps per chunk
#define BLK   256          // threads per heavy block = channels per block
#define CGRP  4            // channel groups (1024/256)
#define EPSV  1e-4f

// ---------------- fast math helpers (gfx1250 v_tanh_f32 / v_rcp_f32) ----------------
__device__ __forceinline__ float fast_rcp(float x) {
#if __has_builtin(__builtin_amdgcn_rcpf)
  return __builtin_amdgcn_rcpf(x);
#else
  return 1.0f / x;
#endif
}

__device__ __forceinline__ float fast_tanh(float x) {
#if __has_builtin(__builtin_amdgcn_tanhf)
  return __builtin_amdgcn_tanhf(x);
#else
  // CDNA5 native transcendental; v_nop covers the TRANS result-use hazard
  // (ISA 7.4: 1 independent op or V_NOP after a TRANS op before output use),
  // since inline asm hides the latency from the scheduler.
  float r;
  asm("v_tanh_f32 %0, %1\n\tv_nop" : "=v"(r) : "v"(x));
  return r;
#endif
}

// sigmoid(x) = 0.5 * (1 + tanh(x/2))
__device__ __forceinline__ float fast_sigmoid(float x) {
  return 0.5f + 0.5f * fast_tanh(0.5f * x);
}

// ---------------- CDNA5 async global -> LDS (ASYNCcnt-tracked data mover) ----------------
// global_load_async_to_lds_b128: VDST = per-lane LDS byte offset, VADDR = 64-bit global VA.
// Tracked by ASYNCcnt; completion is in-order among async loads.
__device__ __forceinline__ void async_load_b128(const float* g, unsigned lds_off) {
  asm volatile("global_load_async_to_lds_b128 %0, %1, off"
               :
               : "v"(lds_off), "v"((unsigned long long)g)
               : "memory");
}

__device__ __forceinline__ unsigned lds_off_of(const void* p) {
  // generic shared pointer: low 32 bits are the workgroup-relative LDS byte offset
  return (unsigned)(unsigned long long)p;
}

// =====================================================================================
// Pass 1: per-chunk product of forget gates f = 1 - sigmoid(g) = 0.5*(1 - tanh(g/2))
// grid (CGRP, NCHUNK, NB) x 256.  Single stream (ig @ +D_H).  NT1=16 rows/tile.
// =====================================================================================
#define NT1 16
#define TILES1 (TCH / NT1)   // 16
__global__ __launch_bounds__(BLK) void glru_pass1_chunkprod(const float* __restrict__ x,
                                                            float* __restrict__ P) {
  __shared__ float buf[2][NT1 * BLK];
  const int tid = threadIdx.x, cg = blockIdx.x, j = blockIdx.y, b = blockIdx.z;
  const int sub = tid >> 6, lane = tid & 63;
  const float* base = x + ((size_t)b * TTOT + (size_t)j * TCH) * CH3 + D_H + cg * BLK;

  auto issue = [&](int k) {
    const float* src = base + (size_t)k * NT1 * CH3;
    unsigned lb = lds_off_of(&buf[k & 1][0]);
#pragma unroll
    for (int r = 0; r < NT1 / 4; ++r) {            // 4 issues / thread / tile
      int t = r * 4 + sub;                         // one slice == one timestep row (1KB)
      async_load_b128(src + (size_t)t * CH3 + lane * 4,
                      lb + (unsigned)((t * BLK + lane * 4) * 4));
    }
  };

  issue(0);
  float prod = 1.0f;
  for (int k = 0; k < TILES1; ++k) {
    if (k + 1 < TILES1) {
      issue(k + 1);
      __builtin_amdgcn_s_wait_asynccnt(4);         // tile k resident, k+1 in flight
    } else {
      __builtin_amdgcn_s_wait_asynccnt(0);
    }
    __syncthreads();
    const float* tb = buf[k & 1];
#pragma unroll
    for (int t = 0; t < NT1; ++t) {
      float g = tb[t * BLK + tid];
      prod *= 0.5f - 0.5f * fast_tanh(0.5f * g);   // f = 1 - sigmoid(g)
    }
    __syncthreads();
  }
  P[((size_t)b * NCHUNK + j) * D_H + cg * BLK + tid] = prod;
}

// =====================================================================================
// Pass 2: per-lane exclusive cumprod over chunk products -> carry-in c for each chunk
// =====================================================================================
__global__ __launch_bounds__(BLK) void glru_pass2_scanprod(const float* __restrict__ P,
                                                           float* __restrict__ CIN) {
  const int L = blockIdx.x * BLK + threadIdx.x;    // 0..8191
  const int b = L >> 10, ch = L & (D_H - 1);
  const float* p = P + (size_t)b * NCHUNK * D_H + ch;
  float* ci = CIN + (size_t)b * NCHUNK * D_H + ch;
  float c = 1.0f;
#pragma unroll
  for (int j = 0; j < NCHUNK; ++j) {
    ci[(size_t)j * D_H] = c;
    c *= p[(size_t)j * D_H];
  }
}

// =====================================================================================
// Pass 3: per-chunk sum of kv/(c+eps).  Streams: input(0), ig(1).  NT3=8 rows/tile.
// =====================================================================================
#define NT3 8
#define TILES3 (TCH / NT3)   // 32
__global__ __launch_bounds__(BLK) void glru_pass3_chunksum(const float* __restrict__ x,
                                                           const float* __restrict__ CIN,
                                                           float* __restrict__ SSUM) {
  __shared__ float buf[2][NT3 * 2 * BLK];
  const int tid = threadIdx.x, cg = blockIdx.x, j = blockIdx.y, b = blockIdx.z;
  const int sub = tid >> 6, lane = tid & 63;
  const float* base = x + ((size_t)b * TTOT + (size_t)j * TCH) * CH3 + cg * BLK;

  auto issue = [&](int k) {
    const float* src = base + (size_t)k * NT3 * CH3;
    unsigned lb = lds_off_of(&buf[k & 1][0]);
#pragma unroll
    for (int r = 0; r < (NT3 * 2) / 4; ++r) {      // 4 issues / thread / tile
      int slice = r * 4 + sub;                     // slice -> (t, stream)
      int t = slice >> 1, s = slice & 1;
      async_load_b128(src + (size_t)t * CH3 + s * D_H + lane * 4,
                      lb + (unsigned)((slice * BLK + lane * 4) * 4));
    }
  };

  float c = CIN[((size_t)b * NCHUNK + j) * D_H + cg * BLK + tid];
  float acc = 0.0f;
  issue(0);
  for (int k = 0; k < TILES3; ++k) {
    if (k + 1 < TILES3) {
      issue(k + 1);
      __builtin_amdgcn_s_wait_asynccnt(4);
    } else {
      __builtin_amdgcn_s_wait_asynccnt(0);
    }
    __syncthreads();
    const float* tb = buf[k & 1];
#pragma unroll
    for (int t = 0; t < NT3; ++t) {
      float xin = tb[(t * 2 + 0) * BLK + tid];
      float gi  = tb[(t * 2 + 1) * BLK + tid];
      float it = fast_tanh(xin);
      float ig = fast_sigmoid(gi);
      c *= (1.0f - ig);                            // running global cumprod
      acc += (it * ig) * fast_rcp(c + EPSV);
    }
    __syncthreads();
  }
  SSUM[((size_t)b * NCHUNK + j) * D_H + cg * BLK + tid] = acc;
}

// =====================================================================================
// Pass 4: per-lane exclusive cumsum over chunk sums -> carry-in S for each chunk
// =====================================================================================
__global__ __launch_bounds__(BLK) void glru_pass4_scansum(const float* __restrict__ SSUM,
                                                          float* __restrict__ SIN) {
  const int L = blockIdx.x * BLK + threadIdx.x;
  const int b = L >> 10, ch = L & (D_H - 1);
  const float* p = SSUM + (size_t)b * NCHUNK * D_H + ch;
  float* si = SIN + (size_t)b * NCHUNK * D_H + ch;
  float s = 0.0f;
#pragma unroll
  for (int j = 0; j < NCHUNK; ++j) {
    si[(size_t)j * D_H] = s;
    s += p[(size_t)j * D_H];
  }
}

// =====================================================================================
// Pass 5: final output.  Streams: input(0), ig(1), og(2).  NT5=8 rows/tile.
// y_t = c_t * (S_in + local cumsum kv/(c+eps));  out = tanh(y) * sigmoid(og)
// =====================================================================================
#define NT5 8
#define TILES5 (TCH / NT5)   // 32
__global__ __launch_bounds__(BLK) void glru_pass5_output(const float* __restrict__ x,
                                                         const float* __restrict__ CIN,
                                                         const float* __restrict__ SIN,
                                                         float* __restrict__ out) {
  __shared__ float buf[2][NT5 * 3 * BLK];
  const int tid = threadIdx.x, cg = blockIdx.x, j = blockIdx.y, b = blockIdx.z;
  const int sub = tid >> 6, lane = tid & 63;
  const float* base = x + ((size_t)b * TTOT + (size_t)j * TCH) * CH3 + cg * BLK;

  auto issue = [&](int k) {
    const float* src = base + (size_t)k * NT5 * CH3;
    unsigned lb = lds_off_of(&buf[k & 1][0]);
#pragma unroll
    for (int r = 0; r < (NT5 * 3) / 4; ++r) {      // 6 issues / thread / tile
      int slice = r * 4 + sub;                     // slice -> (t, stream) over 24 slices
      int t = slice / 3, s = slice % 3;
      async_load_b128(src + (size_t)t * CH3 + s * D_H + lane * 4,
                      lb + (unsigned)((slice * BLK + lane * 4) * 4));
    }
  };

  const size_t lidx = ((size_t)b * NCHUNK + j) * D_H + cg * BLK + tid;
  float c = CIN[lidx];
  float S = SIN[lidx];
  float* obase = out + ((size_t)b * TTOT + (size_t)j * TCH) * D_H + cg * BLK + tid;

  issue(0);
  for (int k = 0; k < TILES5; ++k) {
    if (k + 1 < TILES5) {
      issue(k + 1);
      __builtin_amdgcn_s_wait_asynccnt(6);
    } else {
      __builtin_amdgcn_s_wait_asynccnt(0);
    }
    __syncthreads();
    const float* tb = buf[k & 1];
    float* orow = obase + (size_t)k * NT5 * D_H;
#pragma unroll
    for (int t = 0; t < NT5; ++t) {
      float xin = tb[(t * 3 + 0) * BLK + tid];
      float gi  = tb[(t * 3 + 1) * BLK + tid];
      float go  = tb[(t * 3 + 2) * BLK + tid];
      float it = fast_tanh(xin);
      float ig = fast_sigmoid(gi);
      float og = fast_sigmoid(go);
      c *= (1.0f - ig);
      S += (it * ig) * fast_rcp(c + EPSV);
      float y = c * S;
      orow[(size_t)t * D_H] = fast_tanh(y) * og;
    }
    __syncthreads();
  }
}

// =====================================================================================
extern "C" void kernel_launch(void* const* d_in, const int* in_sizes, int n_in,
                              void* d_out, int out_size, void* d_ws, size_t ws_size,
                              hipStream_t stream) {
  (void)in_sizes; (void)n_in; (void)out_size; (void)ws_size;
  const float* x = (const float*)d_in[0];
  float* out = (float*)d_out;

  // workspace: 4 arrays of NB*NCHUNK*D_H floats (512 KB each, 2 MB total)
  const size_t CHUNK_ELEMS = (size_t)NB * NCHUNK * D_H;   // 131072
  float* w    = (float*)d_ws;
  float* P    = w;
  float* CIN  = w + CHUNK_ELEMS;
  float* SSUM = w + 2 * CHUNK_ELEMS;
  float* SIN  = w + 3 * CHUNK_ELEMS;

  dim3 blk(BLK);
  dim3 grid_h(CGRP, NCHUNK, NB);   // 512 blocks
  dim3 grid_s((NB * D_H) / BLK);   // 32 blocks

  glru_pass1_chunkprod<<<grid_h, blk, 0, stream>>>(x, P);
  glru_pass2_scanprod <<<grid_s, blk, 0, stream>>>(P, CIN);
  glru_pass3_chunksum <<<grid_h, blk, 0, stream>>>(x, CIN, SSUM);
  glru_pass4_scansum  <<<grid_s, blk, 0, stream>>>(SSUM, SIN);
  glru_pass5_output   <<<grid_h, blk, 0, stream>>>(x, CIN, SIN, out);
}